// Clustering_57964878626896
// MI455X (gfx1250) — compile-verified
//
#include <hip/hip_runtime.h>
#include <hip/hip_bf16.h>
#include <math.h>

typedef __attribute__((ext_vector_type(2))) float v2f;
typedef __attribute__((ext_vector_type(8))) float v8f;

#define B_    8
#define H_    16
#define L_    4096
#define D_    64
#define M_    8
#define C_    5
#define DFLAT 8192                    // h*m*d = 16*8*64
#define CTX_ELEMS (33554432UL)        // 8*16*4096*64
#define LSPLIT 16                     // l-dim split for shrink (occupancy)
#define LCHUNK (L_ / LSPLIT)          // 256 l per block
#define WSTR   260                    // padded LDS row stride for W chunk
#define KSTR   68                     // padded LDS row stride for Ks tile

// ---------------------------------------------------------------------------
// Kernel 1: shrink partials.  part[chunk, bh, m, d] =
//     sum_{l in chunk} W[m, l] * X[bh, l, d]
// Grid: (B*H*LSPLIT) blocks x 256 thr (8 waves).  Wave w: d-tile (w&3),
// l-half (w>>2) of the 256-l chunk -> 32 steps of V_WMMA_F32_16X16X4_F32.
// W chunk staged in LDS with rows 8..15 pre-zeroed (so no mask math in the
// hot loop: ds_load_b64 -> v_wmma directly) and bank-padded stride.
// A(16x4): lanes0-15 K=0/1 in VGPR0/1, lanes16-31 K=2/3 (ISA layout);
// B(4x16) mirrored; C/D rows 8..15 are zero padding.
// ---------------------------------------------------------------------------
__global__ void shrink_kernel(const float* __restrict__ X, const float* __restrict__ W,
                              float* __restrict__ part)
{
    __shared__ float wlds[16 * WSTR];            // rows 0..7 real, 8..15 zero
    __shared__ float red[4][32][8];

    const int bh    = blockIdx.x >> 4;           // 0..127
    const int chunk = blockIdx.x & (LSPLIT - 1); // 0..15
    const int t     = threadIdx.x;
    const int wv    = t >> 5, lane = t & 31;
    const int d0    = (wv & 3) * 16;
    const int lh    = wv >> 2;                   // l half within chunk
    const int nrow  = lane & 15;                 // M row (A) == N col (B)
    const bool hi   = lane >= 16;
    const int koff  = hi ? 2 : 0;

    // stage W[0..7][chunk*256 .. +256) into LDS rows 0..7 (coalesced),
    // zero rows 8..15 once
    #pragma unroll
    for (int j = 0; j < M_; ++j) {
        wlds[j * WSTR + t]          = W[(size_t)j * L_ + chunk * LCHUNK + t];
        wlds[(j + M_) * WSTR + t]   = 0.f;
    }
    __syncthreads();

    const float* Xb = X + (size_t)bh * L_ * D_ + (size_t)(chunk * LCHUNK) * D_;
    const int lbase = lh * (LCHUNK / 2);         // 0 or 128 within chunk
    const float* wrow = wlds + nrow * WSTR;      // rows 8..15 read zeros

    v8f acc = {};
    for (int i = 0; i < LCHUNK / 2; i += 4) {
        const int ll = lbase + i + koff;         // l within chunk
        v2f a, b;
        a.x = wrow[ll];
        a.y = wrow[ll + 1];
        b.x = Xb[(size_t)ll * D_ + d0 + nrow];
        b.y = Xb[(size_t)(ll + 1) * D_ + d0 + nrow];
        acc = __builtin_amdgcn_wmma_f32_16x16x4_f32(false, a, false, b,
                                                    (short)0, acc, false, false);
    }

    // reduce the two l-halves (waves 4..7 -> waves 0..3)
    if (lh == 1) {
        #pragma unroll
        for (int r = 0; r < 8; ++r) red[wv & 3][lane][r] = acc[r];
    }
    __syncthreads();
    if (lh == 0 && !hi) {
        float* po = part + ((size_t)(chunk * (B_ * H_) + bh)) * (M_ * D_);
        #pragma unroll
        for (int r = 0; r < M_; ++r)
            po[r * D_ + d0 + lane] = acc[r] + red[wv][lane][r];
    }
}

// ---------------------------------------------------------------------------
// Kernel 1b: deterministic chunk reduction + bias.
// Out[bh, m, d] = bias[m] + sum_chunk part[chunk, bh, m, d]
// ---------------------------------------------------------------------------
__global__ void reduce_kernel(const float* __restrict__ part, const float* __restrict__ bias,
                              float* __restrict__ Out)
{
    const int g = blockIdx.x * 256 + threadIdx.x;   // 0..65535
    float s = bias[(g >> 6) & (M_ - 1)];
    #pragma unroll
    for (int c = 0; c < LSPLIT; ++c)
        s += part[(size_t)c * (B_ * H_ * M_ * D_) + g];
    Out[g] = s;
}

// ---------------------------------------------------------------------------
// Kernel 2: tiny cluster head + loss.  Single block, 8 waves (one per batch).
// ---------------------------------------------------------------------------
__device__ inline void softmax5(const float* x, float* p) {
    float m = x[0];
    for (int c = 1; c < C_; ++c) m = fmaxf(m, x[c]);
    float s = 0.f;
    for (int c = 0; c < C_; ++c) { p[c] = expf(x[c] - m); s += p[c]; }
    float inv = 1.f / s;
    for (int c = 0; c < C_; ++c) p[c] *= inv;
}

__global__ void loss_kernel(const float* __restrict__ Ks, const float* __restrict__ W_pc,
                            const float* __restrict__ b_pc, const float* __restrict__ W_q,
                            const float* __restrict__ b_q, const float* __restrict__ W_k,
                            const float* __restrict__ b_k, float* __restrict__ lossOut,
                            float* __restrict__ misc)
{
    __shared__ float zsh[B_][C_];
    const int t = threadIdx.x;
    const int b = t >> 5, lane = t & 31;

    float part[C_] = {0.f, 0.f, 0.f, 0.f, 0.f};
    const float* fb = Ks + (size_t)b * DFLAT;      // flat = Ks.reshape(b,-1)
    for (int i = lane; i < DFLAT; i += 32) {
        float f = fb[i];
        #pragma unroll
        for (int c = 0; c < C_; ++c) part[c] += f * W_pc[(size_t)c * DFLAT + i];
    }
    #pragma unroll
    for (int c = 0; c < C_; ++c)
        for (int off = 16; off > 0; off >>= 1)
            part[c] += __shfl_down(part[c], off, 32);
    if (lane == 0)
        for (int c = 0; c < C_; ++c)
            zsh[b][c] = fmaxf(part[c] + b_pc[c], 0.f);   // relu
    __syncthreads();

    if (t == 0) {
        float cq[B_][C_], ck[B_][C_];
        for (int bb = 0; bb < B_; ++bb) {
            float lq[C_], lk[C_];
            for (int c = 0; c < C_; ++c) {
                float aq = b_q[c], ak = b_k[c];
                for (int j = 0; j < C_; ++j) {
                    aq += zsh[bb][j] * W_q[c * C_ + j];
                    ak += zsh[bb][j] * W_k[c * C_ + j];
                }
                lq[c] = aq; lk[c] = ak;
            }
            softmax5(lq, cq[bb]);
            softmax5(lk, ck[bb]);
        }
        float mu[C_], sig[C_];
        for (int c = 0; c < C_; ++c) {
            float s = 0.f;
            for (int bb = 0; bb < B_; ++bb) s += cq[bb][c];
            mu[c] = s * (1.f / B_);
        }
        for (int c = 0; c < C_; ++c) {
            float mk = 0.f;
            for (int bb = 0; bb < B_; ++bb) mk += ck[bb][c];
            mk *= (1.f / B_);
            float v = 0.f;
            for (int bb = 0; bb < B_; ++bb) { float d = ck[bb][c] - mk; v += d * d; }
            float sd = sqrtf(v / (B_ - 1));            // ddof=1
            sig[c] = log1pf(expf(sd));                 // softplus
        }
        const float HLOG2PI = 0.91893853320467274f;    // 0.5*log(2*pi)
        float lp = 0.f;
        for (int bb = 0; bb < B_; ++bb)
            for (int c = 0; c < C_; ++c) {
                float z = (ck[bb][c] - mu[c]) / sig[c];
                lp += -0.5f * z * z - logf(sig[c]) - HLOG2PI;
            }
        float loss1 = -lp / (B_ * C_);
        float ce = 0.f;
        for (int bb = 0; bb < B_; ++bb) {
            float m = cq[bb][0];
            for (int c = 1; c < C_; ++c) m = fmaxf(m, cq[bb][c]);
            float s = 0.f;
            for (int c = 0; c < C_; ++c) s += expf(cq[bb][c] - m);
            float lse = m + logf(s);
            float e = 0.f;
            for (int c = 0; c < C_; ++c) e += cq[bb][c] * (cq[bb][c] - lse);
            ce += -e;
        }
        ce *= (1.f / B_);
        lossOut[0] = loss1 + ce;
        int ind0 = 0; float best = cq[0][0];
        for (int c = 1; c < C_; ++c) if (cq[0][c] > best) { best = cq[0][c]; ind0 = c; }
        misc[0] = (ind0 >= 1) ? 1.f : 0.f;             // keep0 has a zero iff inds0>=1
    }
}

// ---------------------------------------------------------------------------
// Kernel 3: scores + dual-variant masked softmax -> attn[16,4096,8].
// One wave per (h, 16-row q tile): 16 WMMA f32 16x16x4 steps over d=64.
// All 8 waves of a block share h (256 tiles/h % 8 == 0), so the Ks[0,h]
// tile is staged once per block in LDS, zero-padded to 16 rows, stride 68.
// ---------------------------------------------------------------------------
__global__ void attn_kernel(const float* __restrict__ Q, const float* __restrict__ Ks,
                            const float* __restrict__ misc, float* __restrict__ attn)
{
    __shared__ float ksh[16 * KSTR];               // rows 0..7 real, 8..15 zero
    __shared__ float sS[8][16][9];                 // [wave][row][k] (+pad)
    const int t = threadIdx.x;
    const int wv = t >> 5, lane = t & 31;
    const int wg = blockIdx.x * 8 + wv;            // 512 blocks * 8 waves = 4096
    const int h = blockIdx.x >> 5;                 // uniform across the block
    const int q0 = (wg & 255) << 4;
    const int nrow = lane & 15;
    const bool hi = lane >= 16;
    const int koff = hi ? 2 : 0;

    const float* Qh = Q + (size_t)h * L_ * D_;     // batch 0 only
    const float* Kh = Ks + (size_t)h * (M_ * D_);  // Ks[0,h]

    // stage Ks tile: 512 real floats + 512 zeros, coalesced
    #pragma unroll
    for (int rep = 0; rep < 2; ++rep) {
        const int e = t + rep * 256;               // 0..511
        ksh[(e >> 6) * KSTR + (e & 63)] = Kh[e];
        const int z = e + 512;                     // 512..1023
        ksh[(z >> 6) * KSTR + (z & 63)] = 0.f;
    }
    __syncthreads();

    const float* krow = ksh + nrow * KSTR;         // rows 8..15 read zeros

    v8f acc = {};
    #pragma unroll 4
    for (int s = 0; s < 16; ++s) {
        const int d0 = s * 4 + koff;
        v2f a, b;
        a.x = Qh[(size_t)(q0 + nrow) * D_ + d0];
        a.y = Qh[(size_t)(q0 + nrow) * D_ + d0 + 1];
        b.x = krow[d0];
        b.y = krow[d0 + 1];
        acc = __builtin_amdgcn_wmma_f32_16x16x4_f32(false, a, false, b,
                                                    (short)0, acc, false, false);
    }
    // scatter tile to LDS: VGPR r -> (row r | r+8, k = lane&15); scale 1/sqrt(64)
    if (nrow < M_) {
        #pragma unroll
        for (int r = 0; r < 8; ++r)
            sS[wv][hi ? r + 8 : r][nrow] = acc[r] * 0.125f;
    }
    __syncthreads();

    const float hasZero = misc[0];
    if (lane < 16) {
        const int q = q0 + lane;
        float xa[M_], xb[M_];
        #pragma unroll
        for (int k = 0; k < M_; ++k) {
            bool masked = (k <= q);                // tril mask -> -1e9
            xa[k] = masked ? -1e9f : sS[wv][lane][k];
            xb[k] = masked ? -1e9f : 0.f;          // zeroed-cluster variant
        }
        float ma = -3e38f, mb = -3e38f;
        #pragma unroll
        for (int k = 0; k < M_; ++k) { ma = fmaxf(ma, xa[k]); mb = fmaxf(mb, xb[k]); }
        float sa = 0.f, sb = 0.f;
        #pragma unroll
        for (int k = 0; k < M_; ++k) {
            xa[k] = expf(xa[k] - ma); sa += xa[k];
            xb[k] = expf(xb[k] - mb); sb += xb[k];
        }
        const float ra = 8.f / sa, rb = 8.f / sb;  // *lk2 after softmax
        float fin[M_], mf = -3e38f;
        #pragma unroll
        for (int k = 0; k < M_; ++k) {
            float f = xa[k] * ra;
            if (hasZero > 0.5f) f = fmaxf(f, xb[k] * rb);   // max over clusters
            fin[k] = f; mf = fmaxf(mf, f);
        }
        float se = 0.f;
        #pragma unroll
        for (int k = 0; k < M_; ++k) { fin[k] = expf(fin[k] - mf); se += fin[k]; }
        const float inv = 1.f / se;
        const size_t base = ((size_t)h * L_ + q) * M_;
        #pragma unroll
        for (int k = 0; k < M_; ++k) attn[base + k] = fin[k] * inv;
    }
}

// ---------------------------------------------------------------------------
// Kernel 4: context[b,h,q,d] = sum_k attn[h,q,k] * Vs[b,h,k,d].  128 MB store-
// bound; Vs tile in LDS, q split 8x across blocks for occupancy, coalesced
// writes (threads 0..63 cover d for one q row).
// ---------------------------------------------------------------------------
__global__ void context_kernel(const float* __restrict__ attn, const float* __restrict__ Vs,
                               float* __restrict__ Out)
{
    __shared__ float vsh[M_ * D_];
    const int bh = blockIdx.x >> 3;                // 0..127
    const int qc = blockIdx.x & 7;                 // q chunk of 512
    const int h = bh & (H_ - 1);
    const int t = threadIdx.x;
    vsh[t]       = Vs[(size_t)bh * (M_ * D_) + t];
    vsh[t + 256] = Vs[(size_t)bh * (M_ * D_) + 256 + t];
    __syncthreads();

    const int d = t & 63;
    const int qoff = t >> 6;
    const float* ah = attn + (size_t)h * L_ * M_;
    float* ob = Out + (size_t)bh * L_ * D_;
    const int qlo = qc * 512;
    for (int q0 = qlo; q0 < qlo + 512; q0 += 4) {
        const int q = q0 + qoff;
        const float* ar = ah + (size_t)q * M_;
        float s = 0.f;
        #pragma unroll
        for (int k = 0; k < M_; ++k) s += ar[k] * vsh[k * D_ + d];
        ob[(size_t)q * D_ + d] = s;
    }
}

// ---------------------------------------------------------------------------
extern "C" void kernel_launch(void* const* d_in, const int* in_sizes, int n_in,
                              void* d_out, int out_size, void* d_ws, size_t ws_size,
                              hipStream_t stream) {
    (void)in_sizes; (void)n_in; (void)out_size; (void)ws_size;
    const float* Q    = (const float*)d_in[0];
    const float* K    = (const float*)d_in[1];
    const float* V    = (const float*)d_in[2];
    const float* W_sk = (const float*)d_in[3];
    const float* b_sk = (const float*)d_in[4];
    const float* W_sv = (const float*)d_in[5];
    const float* b_sv = (const float*)d_in[6];
    const float* W_pc = (const float*)d_in[7];
    const float* b_pc = (const float*)d_in[8];
    const float* W_q  = (const float*)d_in[9];
    const float* b_q  = (const float*)d_in[10];
    const float* W_k  = (const float*)d_in[11];
    const float* b_k  = (const float*)d_in[12];

    float* out = (float*)d_out;                    // context[0..CTX), loss at CTX
    float* ws  = (float*)d_ws;
    float* wsKs   = ws;                            // 65536 floats
    float* wsVs   = ws + 65536;                    // 65536 floats
    float* wsMisc = ws + 131072;                   // 16 floats
    float* wsAttn = ws + 131088;                   // 524288 floats
    float* wsPart = ws + 655376;                   // LSPLIT*65536 = 1048576 floats

    shrink_kernel <<<B_ * H_ * LSPLIT, 256, 0, stream>>>(K, W_sk, wsPart);
    reduce_kernel <<<256, 256, 0, stream>>>(wsPart, b_sk, wsKs);
    shrink_kernel <<<B_ * H_ * LSPLIT, 256, 0, stream>>>(V, W_sv, wsPart);
    reduce_kernel <<<256, 256, 0, stream>>>(wsPart, b_sv, wsVs);
    loss_kernel   <<<1, 256, 0, stream>>>(wsKs, W_pc, b_pc, W_q, b_q, W_k, b_k,
                                          out + CTX_ELEMS, wsMisc);
    attn_kernel   <<<512, 256, 0, stream>>>(Q, wsKs, wsMisc, wsAttn);
    context_kernel<<<B_ * H_ * 8, 256, 0, stream>>>(wsAttn, wsVs, out);
}